// RotaryPositionalEmbedding_84645215470161
// MI455X (gfx1250) — compile-verified
//
#include <hip/hip_runtime.h>
#include <math.h>

// RoPE: out[b,s,2k]   = cos(a)*x[b,s,2k] - sin(a)*x[b,s,2k+1]
//       out[b,s,2k+1] = sin(a)*x[b,s,2k] + cos(a)*x[b,s,2k+1]
// a = pos[s] * 10000^(-2k/128)
//
// Memory-bound: ~17MB traffic -> ~0.75us floor @ 23.3 TB/s.
// Strategy: precompute L2-resident cos/sin table (2MB) once (cold kernel),
// then stream x/out with NT b128 accesses. The table chunk for a block's 8
// positions is staged to LDS via the gfx1250 async-to-LDS path (ASYNCcnt)
// and reused across all 4 batch images (batch is the reuse axis).

#define SEQ     4096
#define DK      128
#define HALF_DK 64
#define BATCH   4
#define F4_PER_BATCH (SEQ * DK / 4)   // 131072 float4s per batch image

typedef float f32x4 __attribute__((ext_vector_type(4)));
typedef float f32x2 __attribute__((ext_vector_type(2)));

// log2(10000)/64
#define LOG2_THETA_OVER_HALFDK 0.20762050593045997f

// ---------------------------------------------------------------------------
// Kernel A (cold): build cos/sin table, float2 {cos,sin} at [pos][k]
// (floats: table[pos*128 + 2k] = cos, table[pos*128 + 2k + 1] = sin),
// matching the even/odd pair layout of x exactly.
// ---------------------------------------------------------------------------
__global__ __launch_bounds__(256) void rope_table_kernel(
    const int* __restrict__ token_positions, float* __restrict__ table) {
  const int tid = blockIdx.x * 256 + threadIdx.x;   // 0 .. SEQ*HALF_DK-1
  const int p = tid >> 6;                            // position index
  const int k = tid & (HALF_DK - 1);                 // frequency index
  const float inv_freq = exp2f(-(float)k * LOG2_THETA_OVER_HALFDK);
  const float angle = (float)token_positions[p] * inv_freq;
  float s, c;
  sincosf(angle, &s, &c);                            // accurate range reduction
  f32x2 cs; cs.x = c; cs.y = s;
  ((f32x2*)table)[tid] = cs;                         // coalesced b64 stores
}

// ---------------------------------------------------------------------------
// Kernel B (hot): streaming apply. 512 blocks, each covers 8 consecutive
// positions. Per thread: one table float4 {c0,s0,c1,s1} (async-staged to
// LDS), applied to the matching float4 of ALL 4 batch images.
// ---------------------------------------------------------------------------
__global__ __launch_bounds__(256) void rope_apply_kernel(
    const float* __restrict__ x, const float* __restrict__ table,
    float* __restrict__ out) {
  __shared__ f32x4 tbl[256];                         // 4KB table slice

  const int tid = threadIdx.x;
  const unsigned blk = blockIdx.x;                   // 0..511
  const unsigned idx = blk * 256u + (unsigned)tid;   // f4 index within a batch

  // --- async-stage this block's table chunk into LDS (gfx1250 ASYNC path) ---
  const f32x4* tsrc = (const f32x4*)table + idx;
  // Generic pointer to __shared__: low 32 bits are the LDS byte offset.
  unsigned lds_addr = (unsigned)(unsigned long long)(&tbl[tid]);
  unsigned long long gaddr = (unsigned long long)tsrc;
  asm volatile("global_load_async_to_lds_b128 %0, %1, off"
               :: "v"(lds_addr), "v"(gaddr) : "memory");

  // --- overlap: issue 4 independent NT b128 loads (one per batch image) ---
  const f32x4* xp = (const f32x4*)x;
  f32x4 xv0 = __builtin_nontemporal_load(xp + idx + 0u * F4_PER_BATCH);
  f32x4 xv1 = __builtin_nontemporal_load(xp + idx + 1u * F4_PER_BATCH);
  f32x4 xv2 = __builtin_nontemporal_load(xp + idx + 2u * F4_PER_BATCH);
  f32x4 xv3 = __builtin_nontemporal_load(xp + idx + 3u * F4_PER_BATCH);

  asm volatile("s_wait_asynccnt 0x0" ::: "memory");
  __syncthreads();

  // --- rotate two (even, odd) pairs per batch image with one table read ---
  const f32x4 t = tbl[tid];                          // {c0,s0,c1,s1}
  f32x4* op = (f32x4*)out;

  f32x4 o0, o1, o2, o3;
  o0.x = t.x * xv0.x - t.y * xv0.y;  o0.y = t.y * xv0.x + t.x * xv0.y;
  o0.z = t.z * xv0.z - t.w * xv0.w;  o0.w = t.w * xv0.z + t.z * xv0.w;

  o1.x = t.x * xv1.x - t.y * xv1.y;  o1.y = t.y * xv1.x + t.x * xv1.y;
  o1.z = t.z * xv1.z - t.w * xv1.w;  o1.w = t.w * xv1.z + t.z * xv1.w;

  o2.x = t.x * xv2.x - t.y * xv2.y;  o2.y = t.y * xv2.x + t.x * xv2.y;
  o2.z = t.z * xv2.z - t.w * xv2.w;  o2.w = t.w * xv2.z + t.z * xv2.w;

  o3.x = t.x * xv3.x - t.y * xv3.y;  o3.y = t.y * xv3.x + t.x * xv3.y;
  o3.z = t.z * xv3.z - t.w * xv3.w;  o3.w = t.w * xv3.z + t.z * xv3.w;

  __builtin_nontemporal_store(o0, op + idx + 0u * F4_PER_BATCH);
  __builtin_nontemporal_store(o1, op + idx + 1u * F4_PER_BATCH);
  __builtin_nontemporal_store(o2, op + idx + 2u * F4_PER_BATCH);
  __builtin_nontemporal_store(o3, op + idx + 3u * F4_PER_BATCH);
}

// ---------------------------------------------------------------------------
// Fallback: fused kernel (no workspace table), sincos inline per thread.
// ---------------------------------------------------------------------------
__global__ __launch_bounds__(256) void rope_fused_kernel(
    const float* __restrict__ x, const int* __restrict__ token_positions,
    float* __restrict__ out) {
  const long long idx = (long long)blockIdx.x * 256 + threadIdx.x; // f4 index
  const int in_batch = (int)(idx & (SEQ * 32 - 1));  // f4 index within batch
  const int p = in_batch >> 5;
  const int tk = in_batch & 31;                       // f4 within row
  const float pos = (float)token_positions[p];
  const float k0 = (float)(2 * tk);
  float s0, c0, s1, c1;
  sincosf(pos * exp2f(-k0 * LOG2_THETA_OVER_HALFDK), &s0, &c0);
  sincosf(pos * exp2f(-(k0 + 1.0f) * LOG2_THETA_OVER_HALFDK), &s1, &c1);
  f32x4 xv = __builtin_nontemporal_load((const f32x4*)x + idx);
  f32x4 o;
  o.x = c0 * xv.x - s0 * xv.y;
  o.y = s0 * xv.x + c0 * xv.y;
  o.z = c1 * xv.z - s1 * xv.w;
  o.w = s1 * xv.z + c1 * xv.w;
  __builtin_nontemporal_store(o, (f32x4*)out + idx);
}

extern "C" void kernel_launch(void* const* d_in, const int* in_sizes, int n_in,
                              void* d_out, int out_size, void* d_ws,
                              size_t ws_size, hipStream_t stream) {
  const float* x = (const float*)d_in[0];
  const int* token_positions = (const int*)d_in[1];
  float* out = (float*)d_out;

  const size_t table_bytes = (size_t)SEQ * DK * sizeof(float);  // 2 MB
  const int table_blocks = SEQ * HALF_DK / 256;                 // 1024
  const int apply_blocks = F4_PER_BATCH / 256;                  // 512

  if (ws_size >= table_bytes) {
    float* table = (float*)d_ws;
    rope_table_kernel<<<table_blocks, 256, 0, stream>>>(token_positions, table);
    rope_apply_kernel<<<apply_blocks, 256, 0, stream>>>(x, table, out);
  } else {
    const int fused_blocks = BATCH * F4_PER_BATCH / 256;        // 2048
    rope_fused_kernel<<<fused_blocks, 256, 0, stream>>>(x, token_positions,
                                                        out);
  }
}